// Method_GCN_4131758539006
// MI455X (gfx1250) — compile-verified
//
#include <hip/hip_runtime.h>
#include <stdint.h>

typedef __attribute__((ext_vector_type(2))) float v2f;
typedef __attribute__((ext_vector_type(8))) float v8f;

#define IN_DIM  10
#define HID_DIM 35
#define OUT_DIM 2

// ---------------------------------------------------------------------------
// Degree / normalization
// ---------------------------------------------------------------------------
__global__ __launch_bounds__(256) void k_fill1(float* __restrict__ deg, int n) {
  int i = blockIdx.x * blockDim.x + threadIdx.x;
  if (i < n) deg[i] = 1.0f;  // self-loop contribution
}

__global__ __launch_bounds__(256) void k_deg(const long long* __restrict__ dst,
                                             float* __restrict__ deg, int e) {
  int i = blockIdx.x * blockDim.x + threadIdx.x;
  if (i < e) atomicAdd(&deg[(int)dst[i]], 1.0f);
}

__global__ __launch_bounds__(256) void k_dinv(float* __restrict__ d, int n) {
  int i = blockIdx.x * blockDim.x + threadIdx.x;
  if (i < n) d[i] = rsqrtf(fmaxf(d[i], 1.0f));
}

// ---------------------------------------------------------------------------
// Dense transform via V_WMMA_F32_16X16X4_F32, K/NCOLS compile-time so the
// k-loop fully unrolls and boundary guards become selects (no exec branches).
// One wave computes a 16(M) x 16(N) tile of O = A(MxK) * W(KxN), K stepped 4.
// A layout (16x4 f32): lanes 0-15 hold K=0,1 (v0,v1); lanes 16-31 hold K=2,3.
// B layout (4x16 f32): lanes 0-15 (N=0..15) hold K=0,1; lanes 16-31 K=2,3.
// C/D layout: VGPR r -> M=r (lanes 0-15) / M=r+8 (lanes 16-31), N = lane&15.
// Requires M % 16 == 0 (N_NODES = 500000 = 31250*16).
// ---------------------------------------------------------------------------
template <int K, int NCOLS>
__global__ __launch_bounds__(256) void k_gemm_wmma(const float* __restrict__ A,
                                                   const float* __restrict__ W,
                                                   float* __restrict__ O,
                                                   int M) {
  constexpr int NT = (NCOLS + 15) / 16;
  int wave = blockIdx.x * (blockDim.x >> 5) + (threadIdx.x >> 5);
  int mt = wave / NT;
  int nt = wave - mt * NT;
  if (mt * 16 >= M) return;  // wave-uniform exit: EXEC all-ones for WMMA

  int lane = threadIdx.x & 31;
  int mn   = lane & 15;   // M index for A lanes, N index for B/D lanes
  int kh   = lane >> 4;   // which K-half this lane holds

  long long row = (long long)mt * 16 + mn;  // always < M
  int  col  = nt * 16 + mn;
  bool cok  = col < NCOLS;
  int  colc = cok ? col : (NCOLS - 1);      // clamped, always in-bounds

  const float* __restrict__ Arow = A + row * K;

  v8f c = {};
  constexpr int KSTEPS = (K + 3) / 4;
#pragma unroll
  for (int kb = 0; kb < KSTEPS; ++kb) {
    const int k0 = kb * 4 + kh * 2;
    // Clamp indices (in-bounds unconditional loads), mask values via select.
    const int  ka0 = (k0     < K) ? k0     : (K - 1);
    const int  ka1 = (k0 + 1 < K) ? k0 + 1 : (K - 1);
    const bool m0  = (k0     < K);
    const bool m1  = (k0 + 1 < K);
    float a0 = Arow[ka0];
    float a1 = Arow[ka1];
    float b0 = W[ka0 * NCOLS + colc];
    float b1 = W[ka1 * NCOLS + colc];
    v2f a, b;
    a.x = m0 ? a0 : 0.0f;
    a.y = m1 ? a1 : 0.0f;
    b.x = (m0 && cok) ? b0 : 0.0f;
    b.y = (m1 && cok) ? b1 : 0.0f;
    c = __builtin_amdgcn_wmma_f32_16x16x4_f32(false, a, false, b,
                                              (short)0, c, false, false);
  }

#pragma unroll
  for (int r = 0; r < 8; ++r) {
    long long rr = (long long)mt * 16 + r + kh * 8;  // always < M
    if (cok) O[rr * NCOLS + col] = c[r];
  }
}

// ---------------------------------------------------------------------------
// Propagation: self-loop init, edge scatter-add, activation, log-softmax
// ---------------------------------------------------------------------------
template <int DIM>
__global__ __launch_bounds__(256) void k_selfloop(const float* __restrict__ t,
                                                  const float* __restrict__ dinv,
                                                  float* __restrict__ p, int n) {
  int i = blockIdx.x * blockDim.x + threadIdx.x;
  if (i < n) {
    float nm = dinv[i] * dinv[i];
    long long base = (long long)i * DIM;
#pragma unroll
    for (int f = 0; f < DIM; ++f) p[base + f] = t[base + f] * nm;
  }
}

template <int DIM>
__global__ __launch_bounds__(256) void k_scatter(const long long* __restrict__ ei,
                                                 long long e,
                                                 const float* __restrict__ dinv,
                                                 const float* __restrict__ t,
                                                 float* __restrict__ p) {
  long long i = (long long)blockIdx.x * blockDim.x + threadIdx.x;
  if (i < e) {
    int s = (int)ei[i];
    int d = (int)ei[e + i];
    const float* __restrict__ ts = t + (long long)s * DIM;
    float* __restrict__       pd = p + (long long)d * DIM;
    __builtin_prefetch(ts, 0, 0);  // global_prefetch_b8: random-gather latency
    float nm = dinv[s] * dinv[d];
#pragma unroll
    for (int f = 0; f < DIM; ++f) atomicAdd(&pd[f], ts[f] * nm);
  }
}

template <int DIM>
__global__ __launch_bounds__(256) void k_relu_bias(float* __restrict__ p,
                                                   const float* __restrict__ b,
                                                   int n) {
  int i = blockIdx.x * blockDim.x + threadIdx.x;
  if (i < n) {
    long long base = (long long)i * DIM;
#pragma unroll
    for (int f = 0; f < DIM; ++f) {
      float v = p[base + f] + b[f];
      p[base + f] = v > 0.0f ? v : 0.0f;
    }
  }
}

__global__ __launch_bounds__(256) void k_logsoftmax2(const float* __restrict__ p,
                                                     const float* __restrict__ b,
                                                     float* __restrict__ o, int n) {
  int i = blockIdx.x * blockDim.x + threadIdx.x;
  if (i < n) {
    float a = p[2 * i]     + b[0];
    float c = p[2 * i + 1] + b[1];
    float m = fmaxf(a, c);
    float l = m + logf(expf(a - m) + expf(c - m));
    o[2 * i]     = a - l;
    o[2 * i + 1] = c - l;
  }
}

// ---------------------------------------------------------------------------
// Launch
// ---------------------------------------------------------------------------
extern "C" void kernel_launch(void* const* d_in, const int* in_sizes, int n_in,
                              void* d_out, int out_size, void* d_ws, size_t ws_size,
                              hipStream_t stream) {
  (void)n_in; (void)out_size; (void)ws_size;
  const float*     x  = (const float*)d_in[0];
  const long long* ei = (const long long*)d_in[1];
  const float*     W1 = (const float*)d_in[2];
  const float*     b1 = (const float*)d_in[3];
  const float*     W2 = (const float*)d_in[4];
  const float*     b2 = (const float*)d_in[5];
  float* out = (float*)d_out;

  const int       N = in_sizes[0] / IN_DIM;
  const long long E = (long long)in_sizes[1] / 2;

  // Scratch layout (256B-aligned slices of d_ws)
  size_t off = 0;
  auto carve = [&](size_t nfloats) {
    float* ptr = (float*)((char*)d_ws + off);
    off += ((nfloats * sizeof(float) + 255) / 256) * 256;
    return ptr;
  };
  float* dinv = carve((size_t)N);            // degree, then rsqrt in place
  float* t1   = carve((size_t)N * HID_DIM);  // x @ W1
  float* p1   = carve((size_t)N * HID_DIM);  // propagated + relu'd hidden
  float* t2   = carve((size_t)N * OUT_DIM);  // h1 @ W2
  float* p2   = carve((size_t)N * OUT_DIM);  // propagated logits

  const int TB = 256;
  const int gn = (N + TB - 1) / TB;
  const int ge = (int)((E + TB - 1) / TB);

  // Degree + symmetric normalization
  k_fill1<<<gn, TB, 0, stream>>>(dinv, N);
  k_deg<<<ge, TB, 0, stream>>>(ei + E, dinv, (int)E);
  k_dinv<<<gn, TB, 0, stream>>>(dinv, N);

  // Layer 1: dense transform (WMMA) + propagate + bias/ReLU
  {
    constexpr int NT = (HID_DIM + 15) / 16;  // 3 column tiles
    long long waves = (long long)((N + 15) / 16) * NT;
    int blocks = (int)((waves + 7) / 8);     // 8 waves / 256-thread block
    k_gemm_wmma<IN_DIM, HID_DIM><<<blocks, TB, 0, stream>>>(x, W1, t1, N);
  }
  k_selfloop<HID_DIM><<<gn, TB, 0, stream>>>(t1, dinv, p1, N);
  k_scatter<HID_DIM><<<ge, TB, 0, stream>>>(ei, E, dinv, t1, p1);
  k_relu_bias<HID_DIM><<<gn, TB, 0, stream>>>(p1, b1, N);

  // Layer 2: dense transform (WMMA) + propagate + log-softmax
  {
    constexpr int NT = (OUT_DIM + 15) / 16;  // 1 column tile
    long long waves = (long long)((N + 15) / 16) * NT;
    int blocks = (int)((waves + 7) / 8);
    k_gemm_wmma<HID_DIM, OUT_DIM><<<blocks, TB, 0, stream>>>(p1, W2, t2, N);
  }
  k_selfloop<OUT_DIM><<<gn, TB, 0, stream>>>(t2, dinv, p2, N);
  k_scatter<OUT_DIM><<<ge, TB, 0, stream>>>(ei, E, dinv, t2, p2);
  k_logsoftmax2<<<gn, TB, 0, stream>>>(p2, b2, out, N);
}